// GAT_5652176961988
// MI455X (gfx1250) — compile-verified
//
#include <hip/hip_runtime.h>
#include <math.h>

// ---------------------------------------------------------------------------
// 2-layer GAT for MI455X (gfx1250, wave32).
//   GEMM via V_WMMA_F32_16X16X4_F32 (full f32 precision tensor path).
//   Edge softmax/aggregation via float global atomics; feat (51MB) is L2-resident.
// ---------------------------------------------------------------------------

#define N_NODES 50000
#define N_EDGES 800000
#define IN_DIM 256
#define HID 64
#define HEADS 4
#define FTOT 256          // HEADS*HID
#define NEG_SLOPE 0.2f

typedef __attribute__((ext_vector_type(2))) float v2f;
typedef __attribute__((ext_vector_type(8))) float v8f;

// ---------------------------------------------------------------------------
// feat[N,256] = A[N,256] @ W[256,256]   (f32 WMMA 16x16x4)
// Block: 256 threads = 8 waves. grid = (N/16, 2).
// Wave w computes row-tile blockIdx.x, col-tile blockIdx.y*8 + w.
// A strip (16 rows x 256 cols, contiguous) staged in LDS once per block.
//
// Fragment layouts (CDNA5 ISA 7.12.2):
//  A 16x4 : lanes 0-15 -> M=lane, VGPR0/1 = K0/K1 ; lanes 16-31 -> K2/K3
//  B 4x16 : lanes 0-15 -> N=lane, VGPR0/1 = K0/K1 ; lanes 16-31 -> K2/K3
//  C/D    : VGPR j, lanes 0-15 -> M=j, N=lane ; lanes 16-31 -> M=j+8, N=lane-16
// ---------------------------------------------------------------------------
__global__ __launch_bounds__(256) void gat_gemm_f32_wmma(
    const float* __restrict__ A, const float* __restrict__ W,
    float* __restrict__ out)
{
    __shared__ float lds[16 * 256];     // 16 KB A strip
    const int tid  = threadIdx.x;
    const int row0 = blockIdx.x * 16;

    const float* Ablk = A + (size_t)row0 * FTOT;   // 16x256 strip is contiguous
    #pragma unroll
    for (int i = 0; i < 16; ++i)
        lds[tid + i * 256] = Ablk[tid + i * 256];
    __syncthreads();

    const int wave  = tid >> 5;
    const int lane  = tid & 31;
    const int col0  = (blockIdx.y * 8 + wave) * 16;
    const int hlane = lane & 15;            // M (for A) / N (for B) within tile
    const int koff  = (lane >> 4) * 2;      // 0 for lanes 0-15, 2 for 16-31

    v8f acc = {};
    const float* Wc = W + col0 + hlane;     // column base for this lane
    #pragma unroll 4
    for (int k = 0; k < IN_DIM; k += 4) {
        // A fragment: row = hlane, Ks = k+koff, k+koff+1 (contiguous, 8B aligned)
        v2f a = *(const v2f*)&lds[hlane * 256 + k + koff];
        // B fragment: col = hlane, rows k+koff, k+koff+1 of row-major W
        v2f b;
        b.x = Wc[(size_t)(k + koff) * FTOT];
        b.y = Wc[(size_t)(k + koff + 1) * FTOT];
        acc = __builtin_amdgcn_wmma_f32_16x16x4_f32(
                  false, a, false, b, (short)0, acc, false, false);
    }

    // store: VGPR j -> row = row0 + j + (lane>=16 ? 8 : 0), col = col0 + hlane
    const int srow = row0 + ((lane >> 4) * 8);
    float* Orow = out + (size_t)srow * FTOT + col0 + hlane;
    #pragma unroll
    for (int j = 0; j < 8; ++j)
        Orow[(size_t)j * FTOT] = acc[j];
}

// ---------------------------------------------------------------------------
// el[n,h] = sum_d feat[n,h,d]*al[h,d] ; er likewise. One node per 256-thr block.
// ---------------------------------------------------------------------------
__global__ __launch_bounds__(256) void gat_eler(
    const float* __restrict__ feat, const float* __restrict__ al,
    const float* __restrict__ ar, float* __restrict__ el, float* __restrict__ er)
{
    __shared__ float sl[256], sr[256];
    const int n = blockIdx.x, c = threadIdx.x;
    float v = feat[(size_t)n * FTOT + c];
    sl[c] = v * al[c];
    sr[c] = v * ar[c];
    __syncthreads();
    const int d = c & 63;
    for (int s = 32; s > 0; s >>= 1) {
        if (d < s) { sl[c] += sl[c + s]; sr[c] += sr[c + s]; }
        __syncthreads();
    }
    if (d == 0) {
        el[n * HEADS + (c >> 6)] = sl[c];
        er[n * HEADS + (c >> 6)] = sr[c];
    }
}

// ---------------------------------------------------------------------------
// Init per-layer accumulators: agg=0, m=-inf, denom=0
// ---------------------------------------------------------------------------
__global__ void gat_init(float* __restrict__ agg, float* __restrict__ m,
                         float* __restrict__ denom)
{
    int i = blockIdx.x * blockDim.x + threadIdx.x;
    if (i < N_NODES * FTOT) agg[i] = 0.f;
    if (i < N_NODES * HEADS) { m[i] = -__builtin_inff(); denom[i] = 0.f; }
}

// float atomic max via ordered-int trick (safe on gfx1250)
__device__ __forceinline__ void atomicMaxF(float* addr, float val)
{
    if (val >= 0.f)
        atomicMax((int*)addr, __float_as_int(val));
    else
        atomicMin((unsigned int*)addr, __float_as_uint(val));
}

__device__ __forceinline__ float leaky(float x)
{
    return x > 0.f ? x : NEG_SLOPE * x;
}

// ---------------------------------------------------------------------------
// segment-max of edge scores into m[dst,h]
// ---------------------------------------------------------------------------
__global__ void gat_edge_max(const int* __restrict__ src, const int* __restrict__ dst,
                             const float* __restrict__ el, const float* __restrict__ er,
                             float* __restrict__ m)
{
    int i = blockIdx.x * blockDim.x + threadIdx.x;
    if (i >= N_EDGES * HEADS) return;
    int e = i >> 2, h = i & 3;
    int s = src[e], d = dst[e];
    float x = leaky(el[s * HEADS + h] + er[d * HEADS + h]);
    atomicMaxF(&m[d * HEADS + h], x);
}

// ---------------------------------------------------------------------------
// segment-sum of exp(e - m[dst]) into denom[dst,h]
// ---------------------------------------------------------------------------
__global__ void gat_edge_denom(const int* __restrict__ src, const int* __restrict__ dst,
                               const float* __restrict__ el, const float* __restrict__ er,
                               const float* __restrict__ m, float* __restrict__ denom)
{
    int i = blockIdx.x * blockDim.x + threadIdx.x;
    if (i >= N_EDGES * HEADS) return;
    int e = i >> 2, h = i & 3;
    int s = src[e], d = dst[e];
    float x = leaky(el[s * HEADS + h] + er[d * HEADS + h]);
    atomicAdd(&denom[d * HEADS + h], expf(x - m[d * HEADS + h]));
}

// ---------------------------------------------------------------------------
// agg[dst, :] += alpha[e,h] * feat[src, :]. One edge per 256-thread block,
// one feature element per lane -> coalesced 1KB row gather + f32 atomic adds.
// feat & agg are L2-resident (51MB each vs 192MB L2).
// ---------------------------------------------------------------------------
__global__ __launch_bounds__(256) void gat_edge_scatter(
    const int* __restrict__ src, const int* __restrict__ dst,
    const float* __restrict__ el, const float* __restrict__ er,
    const float* __restrict__ m, const float* __restrict__ denom,
    const float* __restrict__ feat, float* __restrict__ agg)
{
    int e = blockIdx.x;
    int c = threadIdx.x, h = c >> 6;
    int s = src[e], d = dst[e];
    float x = leaky(el[s * HEADS + h] + er[d * HEADS + h]);
    float alpha = expf(x - m[d * HEADS + h]) / denom[d * HEADS + h];
    atomicAdd(&agg[(size_t)d * FTOT + c], feat[(size_t)s * FTOT + c] * alpha);
}

__device__ __forceinline__ float elu1(float x)
{
    return x > 0.f ? x : expf(x) - 1.f;
}

// layer-1 epilogue: in-place bias + ELU (agg becomes h1)
__global__ void gat_finalize_inplace(float* __restrict__ agg,
                                     const float* __restrict__ bias)
{
    int i = blockIdx.x * blockDim.x + threadIdx.x;
    if (i >= N_NODES * FTOT) return;
    agg[i] = elu1(agg[i] + bias[i & 255]);
}

// layer-2 epilogue: bias + ELU, permuted to head-major tuple layout:
// out[h*N*HID + n*HID + d]
__global__ void gat_finalize_out(const float* __restrict__ agg,
                                 const float* __restrict__ bias,
                                 float* __restrict__ out)
{
    int i = blockIdx.x * blockDim.x + threadIdx.x;
    if (i >= N_NODES * FTOT) return;
    int c = i & 255, n = i >> 8;
    float x = elu1(agg[i] + bias[c]);
    int h = c >> 6, d = c & 63;
    out[(size_t)h * (N_NODES * HID) + (size_t)n * HID + d] = x;
}

// ---------------------------------------------------------------------------
extern "C" void kernel_launch(void* const* d_in, const int* in_sizes, int n_in,
                              void* d_out, int out_size, void* d_ws, size_t ws_size,
                              hipStream_t stream)
{
    (void)in_sizes; (void)n_in; (void)out_size; (void)ws_size;

    const float* x   = (const float*)d_in[0];
    const int*   src = (const int*)d_in[1];
    const int*   dst = (const int*)d_in[2];
    const float* W1  = (const float*)d_in[3];
    const float* al1 = (const float*)d_in[4];
    const float* ar1 = (const float*)d_in[5];
    const float* b1  = (const float*)d_in[6];
    const float* W2  = (const float*)d_in[7];
    const float* al2 = (const float*)d_in[8];
    const float* ar2 = (const float*)d_in[9];
    const float* b2  = (const float*)d_in[10];
    float* out = (float*)d_out;

    // workspace carve-up (~105.6 MB)
    char* ws = (char*)d_ws;
    const size_t featB = (size_t)N_NODES * FTOT * sizeof(float);   // 51.2 MB
    const size_t nhB   = (size_t)N_NODES * HEADS * sizeof(float);  // 0.8 MB
    float* feat  = (float*)ws; ws += featB;
    float* agg   = (float*)ws; ws += featB;   // doubles as h1 between layers
    float* el    = (float*)ws; ws += nhB;
    float* er    = (float*)ws; ws += nhB;
    float* m     = (float*)ws; ws += nhB;
    float* denom = (float*)ws; ws += nhB;

    dim3 gGemm(N_NODES / 16, 2), bGemm(256);
    const int NC  = N_NODES * FTOT;
    const int EH  = N_EDGES * HEADS;
    const int gNC = (NC + 255) / 256;
    const int gEH = (EH + 255) / 256;

    // ---------------- layer 1 ----------------
    gat_gemm_f32_wmma<<<gGemm, bGemm, 0, stream>>>(x, W1, feat);
    gat_eler<<<N_NODES, 256, 0, stream>>>(feat, al1, ar1, el, er);
    gat_init<<<gNC, 256, 0, stream>>>(agg, m, denom);
    gat_edge_max<<<gEH, 256, 0, stream>>>(src, dst, el, er, m);
    gat_edge_denom<<<gEH, 256, 0, stream>>>(src, dst, el, er, m, denom);
    gat_edge_scatter<<<N_EDGES, 256, 0, stream>>>(src, dst, el, er, m, denom, feat, agg);
    gat_finalize_inplace<<<gNC, 256, 0, stream>>>(agg, b1);   // agg == h1 now

    // ---------------- layer 2 ----------------
    gat_gemm_f32_wmma<<<gGemm, bGemm, 0, stream>>>(agg, W2, feat);
    // h1 fully consumed by the GEMM above (stream-ordered) -> safe to reuse agg
    gat_eler<<<N_NODES, 256, 0, stream>>>(feat, al2, ar2, el, er);
    gat_init<<<gNC, 256, 0, stream>>>(agg, m, denom);
    gat_edge_max<<<gEH, 256, 0, stream>>>(src, dst, el, er, m);
    gat_edge_denom<<<gEH, 256, 0, stream>>>(src, dst, el, er, m, denom);
    gat_edge_scatter<<<N_EDGES, 256, 0, stream>>>(src, dst, el, er, m, denom, feat, agg);
    gat_finalize_out<<<gNC, 256, 0, stream>>>(agg, b2, out);
}